// Transformer_7584912245136
// MI455X (gfx1250) — compile-verified
//
#include <hip/hip_runtime.h>
#include <hip/hip_bf16.h>

// ---- problem constants (match reference) ----
#define BATCH   4
#define SEQ     4096
#define NH      32
#define HD      128
#define DIM     4096
#define INTER   11008
#define HC      32
#define HEAVY   256
#define QKVDIM  (3*DIM)
#define EPS     1e-5f

typedef float v2f __attribute__((ext_vector_type(2)));
typedef float v8f __attribute__((ext_vector_type(8)));
typedef int   v4i __attribute__((ext_vector_type(4)));

// -------------------------------------------------------------------------
// Async global->LDS copy (gfx1250 GLOBAL_LOAD_ASYNC_TO_LDS_B128) with a
// portable fallback through VGPRs if the builtin isn't declared.
// Builtin prototype (per hipcc diagnostic): (v4i AS1*, v4i AS3*, Ii, Ii)
// -------------------------------------------------------------------------
#if defined(__has_builtin)
#if __has_builtin(__builtin_amdgcn_global_load_async_to_lds_b128)
#define HAVE_ASYNC_LDS 1
#endif
#endif

__device__ __forceinline__ void copy16_g2l(const float* gp, float* lp) {
#ifdef HAVE_ASYNC_LDS
    __builtin_amdgcn_global_load_async_to_lds_b128(
        (__attribute__((address_space(1))) v4i*)gp,
        (__attribute__((address_space(3))) v4i*)lp, 0, 0);
#else
    *(float4*)lp = *(const float4*)gp;
#endif
}

__device__ __forceinline__ void wait_async() {
#ifdef HAVE_ASYNC_LDS
#if __has_builtin(__builtin_amdgcn_s_wait_asynccnt)
    __builtin_amdgcn_s_wait_asynccnt(0);
#else
    asm volatile("s_wait_asynccnt 0" ::: "memory");
#endif
#endif
}

__device__ __forceinline__ v8f wmma_f32(v2f a, v2f b, v8f c) {
    return __builtin_amdgcn_wmma_f32_16x16x4_f32(false, a, false, b,
                                                 (short)0, c, false, false);
}

// -------------------------------------------------------------------------
// RMSNorm: dst[b,i] = src[b,i] * rsqrt(mean(src^2)+eps) * w[i]
// -------------------------------------------------------------------------
__global__ __launch_bounds__(256) void rmsnorm_kernel(
    const float* __restrict__ src, const float* __restrict__ w,
    float* __restrict__ dst)
{
    __shared__ float red[256];
    const int b = blockIdx.x, tid = threadIdx.x;
    const float* s = src + (size_t)b * DIM;
    float ss = 0.f;
    for (int i = tid; i < DIM; i += 256) { float v = s[i]; ss += v * v; }
    red[tid] = ss; __syncthreads();
    for (int off = 128; off > 0; off >>= 1) {
        if (tid < off) red[tid] += red[tid + off];
        __syncthreads();
    }
    const float rs = rsqrtf(red[0] / (float)DIM + EPS);
    float* d = dst + (size_t)b * DIM;
    for (int i = tid; i < DIM; i += 256) d[i] = s[i] * rs * w[i];
}

// -------------------------------------------------------------------------
// WMMA fp32 GEMV, LDS-staged, double-buffered:
//   out[m,n] = sum_k xv[m,k]*W[n,k] (+ res[m,n]); m=0..3 (padded to 16)
// grid.x = N/16, block = 256 (8 waves).  Per tile: block cooperatively
// async-copies a 16 x KT fp32 W tile into LDS (coalesced b128), each wave
// runs WMMAs over its KT/8 k-slice from LDS (stride KT+4 -> conflict-free
// b64 reads), accumulators combined across waves via LDS at the end.
// -------------------------------------------------------------------------
template<int KT>
__global__ __launch_bounds__(256) void gemv_wmma_kernel(
    const float* __restrict__ xv, const float* __restrict__ W,
    float* __restrict__ out, const float* __restrict__ res, int K, int N)
{
    constexpr int LSTR = KT + 4;          // float stride per row (bank skew)
    constexpr int F4PR = KT / 4;          // float4 per row
    constexpr int LD_PER_THR = 16 * KT / 4 / 256;
    constexpr int KTS = KT / 8;           // k-slice per wave

    __shared__ float tile[2][16 * LSTR];
    __shared__ float red[8][4][16];

    const int tid = threadIdx.x, wave = tid >> 5, lane = tid & 31;
    const int n0 = blockIdx.x * 16;
    const int m = lane & 15, half = lane >> 4;
    const float* ap = xv + (size_t)(m & 3) * K;   // pad rows replicate (harmless)
    const int ntiles = K / KT;

    auto prefetch = [&](int buf, int t) {
        const int k0 = t * KT;
        #pragma unroll
        for (int i = 0; i < LD_PER_THR; ++i) {
            const int f = tid + 256 * i;
            const int row = f / F4PR, c = (f % F4PR) * 4;
            copy16_g2l(W + (size_t)(n0 + row) * K + k0 + c,
                       &tile[buf][row * LSTR + c]);
        }
    };

    v8f c0 = {}, c1 = {};
    prefetch(0, 0);
    for (int t = 0; t < ntiles; ++t) {
        const int buf = t & 1;
        wait_async();
        __syncthreads();
        if (t + 1 < ntiles) prefetch(buf ^ 1, t + 1);
        const int k0 = t * KT;
        const float* bl = &tile[buf][m * LSTR];
        const int kb = wave * KTS;
        #pragma unroll
        for (int kk = kb; kk < kb + KTS; kk += 8) {
            v2f a0 = *(const v2f*)(ap + k0 + kk + 2 * half);
            v2f b0 = *(const v2f*)(bl + kk + 2 * half);
            v2f a1 = *(const v2f*)(ap + k0 + kk + 4 + 2 * half);
            v2f b1 = *(const v2f*)(bl + kk + 4 + 2 * half);
            c0 = wmma_f32(a0, b0, c0);
            c1 = wmma_f32(a1, b1, c1);
        }
    }
    __syncthreads();
    if (lane < 16) {
        #pragma unroll
        for (int v = 0; v < 4; ++v) red[wave][v][lane] = c0[v] + c1[v];
    }
    __syncthreads();
    if (tid < 64) {
        const int mm = tid >> 4, nn = tid & 15, n = n0 + nn;
        float s = 0.f;
        #pragma unroll
        for (int w = 0; w < 8; ++w) s += red[w][mm][nn];
        if (res) s += res[(size_t)mm * N + n];
        out[(size_t)mm * N + n] = s;
    }
}

// -------------------------------------------------------------------------
// Fused W1 || W3 GEMV + SiLU*mul:  act[m,n] = silu(x@W1^T) * (x@W3^T)
// Same staging scheme, two weight streams (KT=128 keeps LDS ~34KB).
// -------------------------------------------------------------------------
template<int KT>
__global__ __launch_bounds__(256) void gemv2_silu_kernel(
    const float* __restrict__ xv, const float* __restrict__ W1,
    const float* __restrict__ W3, float* __restrict__ out, int K, int N)
{
    constexpr int LSTR = KT + 4;
    constexpr int F4PR = KT / 4;
    constexpr int LD_PER_THR = 16 * KT / 4 / 256;
    constexpr int KTS = KT / 8;

    __shared__ float t1[2][16 * LSTR];
    __shared__ float t3[2][16 * LSTR];
    __shared__ float r1[8][4][16];
    __shared__ float r3[8][4][16];

    const int tid = threadIdx.x, wave = tid >> 5, lane = tid & 31;
    const int n0 = blockIdx.x * 16;
    const int m = lane & 15, half = lane >> 4;
    const float* ap = xv + (size_t)(m & 3) * K;
    const int ntiles = K / KT;

    auto prefetch = [&](int buf, int t) {
        const int k0 = t * KT;
        #pragma unroll
        for (int i = 0; i < LD_PER_THR; ++i) {
            const int f = tid + 256 * i;
            const int row = f / F4PR, c = (f % F4PR) * 4;
            copy16_g2l(W1 + (size_t)(n0 + row) * K + k0 + c,
                       &t1[buf][row * LSTR + c]);
            copy16_g2l(W3 + (size_t)(n0 + row) * K + k0 + c,
                       &t3[buf][row * LSTR + c]);
        }
    };

    v8f a1acc = {}, a3acc = {};
    prefetch(0, 0);
    for (int t = 0; t < ntiles; ++t) {
        const int buf = t & 1;
        wait_async();
        __syncthreads();
        if (t + 1 < ntiles) prefetch(buf ^ 1, t + 1);
        const int k0 = t * KT;
        const float* b1l = &t1[buf][m * LSTR];
        const float* b3l = &t3[buf][m * LSTR];
        const int kb = wave * KTS;
        #pragma unroll
        for (int kk = kb; kk < kb + KTS; kk += 4) {
            v2f a  = *(const v2f*)(ap + k0 + kk + 2 * half);
            v2f b1 = *(const v2f*)(b1l + kk + 2 * half);
            v2f b3 = *(const v2f*)(b3l + kk + 2 * half);
            a1acc = wmma_f32(a, b1, a1acc);
            a3acc = wmma_f32(a, b3, a3acc);
        }
    }
    __syncthreads();
    if (lane < 16) {
        #pragma unroll
        for (int v = 0; v < 4; ++v) { r1[wave][v][lane] = a1acc[v]; r3[wave][v][lane] = a3acc[v]; }
    }
    __syncthreads();
    if (tid < 64) {
        const int mm = tid >> 4, nn = tid & 15, n = n0 + nn;
        float g1 = 0.f, g3 = 0.f;
        #pragma unroll
        for (int w = 0; w < 8; ++w) { g1 += r1[w][mm][nn]; g3 += r3[w][mm][nn]; }
        const float silu = g1 / (1.f + __expf(-g1));
        out[(size_t)mm * N + n] = silu * g3;
    }
}

// -------------------------------------------------------------------------
// RoPE on q,k + gather label channels. grid = B*NH, block = 128
// -------------------------------------------------------------------------
__global__ __launch_bounds__(128) void rope_kernel(
    const float* __restrict__ qkv, const float* __restrict__ freqs,
    const int* __restrict__ sorted_channel,
    float* __restrict__ q_rot, float* __restrict__ k_rot,
    float* __restrict__ q_lab, float* __restrict__ k_lab)
{
    __shared__ float qb[HD], kb[HD];
    const int bh = blockIdx.x, b = bh >> 5, h = bh & 31, d = threadIdx.x;
    const float* qs = qkv + (size_t)b * QKVDIM + h * HD;
    const float* ks = qs + DIM;
    const int i = d >> 1;
    const float co = freqs[2 * i], si = freqs[2 * i + 1];
    const float q0 = qs[d & ~1], q1 = qs[d | 1];
    const float k0 = ks[d & ~1], k1 = ks[d | 1];
    const float qr = (d & 1) ? (q1 * co + q0 * si) : (q0 * co - q1 * si);
    const float kr = (d & 1) ? (k1 * co + k0 * si) : (k0 * co - k1 * si);
    qb[d] = qr; kb[d] = kr;
    q_rot[(size_t)bh * HD + d] = qr;
    k_rot[(size_t)bh * HD + d] = kr;
    __syncthreads();
    if (d < HC) {
        const int ch = sorted_channel[h * HC + d];
        q_lab[(size_t)bh * HC + d] = qb[ch];
        k_lab[(size_t)bh * HC + d] = kb[ch];
    }
}

// -------------------------------------------------------------------------
// Label scores + exact top-256 radix-select + sparse attention.
// grid = B*NH, block = 256.  Cache entries at s==pos are overridden with the
// freshly computed k/v (the reference writes the cache; we must not).
// -------------------------------------------------------------------------
__device__ __forceinline__ unsigned fkey(float f) {
    unsigned u = __float_as_uint(f);
    return (u & 0x80000000u) ? ~u : (u | 0x80000000u);   // monotonic key
}

__global__ __launch_bounds__(256) void attn_topk_kernel(
    const float* __restrict__ q_rot, const float* __restrict__ k_rot,
    const float* __restrict__ qkv,   const float* __restrict__ q_lab,
    const float* __restrict__ k_lab, const float* __restrict__ k_label_cache,
    const float* __restrict__ mask1, const float* __restrict__ k_cache,
    const float* __restrict__ v_cache, const int* __restrict__ pos_p,
    float* __restrict__ o_out)
{
    __shared__ float    sc[SEQ];
    __shared__ unsigned hist[256];
    __shared__ unsigned hist2[256];
    __shared__ int      idxbuf[HEAVY];
    __shared__ float    pbuf[HEAVY];
    __shared__ float    qv[HD];
    __shared__ float    qlab[HC];
    __shared__ float    red[256];
    __shared__ unsigned ctr;
    __shared__ int      s_thr;
    __shared__ unsigned s_above;

    const int bh = blockIdx.x, b = bh >> 5, h = bh & 31, tid = threadIdx.x;
    const int pos = *pos_p;

    if (tid < HC) qlab[tid] = q_lab[(size_t)bh * HC + tid];
    if (tid < HD) qv[tid]   = q_rot[(size_t)bh * HD + tid];
    hist[tid] = 0; hist2[tid] = 0;
    if (tid == 0) ctr = 0;
    __syncthreads();

    // ---- label scores ----
    for (int s = tid; s < SEQ; s += 256) {
        const float* kl = (s == pos) ? (k_lab + (size_t)bh * HC)
                                     : (k_label_cache + (((size_t)b * SEQ + s) * NH + h) * HC);
        float acc = 0.f;
        #pragma unroll
        for (int c = 0; c < HC; ++c) acc += qlab[c] * kl[c];
        sc[s] = acc + mask1[s];
    }
    __syncthreads();

    // ---- radix select: level 1 (top 8 bits) ----
    for (int s = tid; s < SEQ; s += 256) atomicAdd(&hist[fkey(sc[s]) >> 24], 1u);
    __syncthreads();
    if (tid == 0) {
        unsigned cum = 0;
        for (int bin = 255; bin >= 0; --bin) {
            const unsigned nc = cum + hist[bin];
            if (nc >= HEAVY) { s_thr = bin; s_above = cum; break; }
            cum = nc;
        }
    }
    __syncthreads();
    const int b1 = s_thr; const unsigned above1 = s_above;

    // ---- level 2 (next 8 bits within boundary bin) ----
    for (int s = tid; s < SEQ; s += 256) {
        const unsigned u = fkey(sc[s]);
        if ((int)(u >> 24) == b1) atomicAdd(&hist2[(u >> 16) & 255u], 1u);
    }
    __syncthreads();
    if (tid == 0) {
        unsigned cum = above1;
        for (int bin = 255; bin >= 0; --bin) {
            const unsigned nc = cum + hist2[bin];
            if (nc >= HEAVY) { s_thr = ((b1 << 8) | bin); s_above = cum; break; }
            cum = nc;
        }
    }
    __syncthreads();
    const unsigned pthr = (unsigned)s_thr;

    // ---- collect indices: strictly greater first, ties fill the rest ----
    for (int s = tid; s < SEQ; s += 256) {
        if ((fkey(sc[s]) >> 16) > pthr) idxbuf[atomicAdd(&ctr, 1u)] = s;
    }
    __syncthreads();
    for (int s = tid; s < SEQ; s += 256) {
        if ((fkey(sc[s]) >> 16) == pthr) {
            const unsigned slot = atomicAdd(&ctr, 1u);
            if (slot < HEAVY) idxbuf[slot] = s;
        }
    }
    __syncthreads();

    // ---- attention scores over selected tokens ----
    const float* knew = k_rot + (size_t)bh * HD;
    const float* vnew = qkv + (size_t)b * QKVDIM + 2 * DIM + h * HD;
    {
        const int j = tid, sidx = idxbuf[j];
        const float* kp = (sidx == pos) ? knew
                        : (k_cache + (((size_t)b * SEQ + sidx) * NH + h) * HD);
        float a = 0.f;
        #pragma unroll 4
        for (int d = 0; d < HD; ++d) a += qv[d] * kp[d];
        a *= 0.08838834764831845f;           // 1/sqrt(128)
        pbuf[j] = a; red[j] = a;
    }
    __syncthreads();
    for (int off = 128; off > 0; off >>= 1) {
        if (tid < off) red[tid] = fmaxf(red[tid], red[tid + off]);
        __syncthreads();
    }
    const float mx = red[0]; __syncthreads();
    const float e = __expf(pbuf[tid] - mx);
    pbuf[tid] = e; red[tid] = e; __syncthreads();
    for (int off = 128; off > 0; off >>= 1) {
        if (tid < off) red[tid] += red[tid + off];
        __syncthreads();
    }
    const float inv_sum = 1.f / red[0]; __syncthreads();

    // ---- o = sum_j p_j * v_sel[j]  (split j range across block halves) ----
    {
        const int d = tid & (HD - 1), jbeg = (tid >> 7) * 128;
        float acc = 0.f;
        for (int jj = 0; jj < 128; ++jj) {
            const int j = jbeg + jj, sidx = idxbuf[j];
            const float* vp = (sidx == pos) ? vnew
                            : (v_cache + (((size_t)b * SEQ + sidx) * NH + h) * HD);
            acc += pbuf[j] * vp[d];
        }
        red[tid] = acc;
    }
    __syncthreads();
    if (tid < HD) o_out[(size_t)bh * HD + tid] = (red[tid] + red[tid + 128]) * inv_sum;
}

// -------------------------------------------------------------------------
extern "C" void kernel_launch(void* const* d_in, const int* in_sizes, int n_in,
                              void* d_out, int out_size, void* d_ws, size_t ws_size,
                              hipStream_t stream) {
    const float* x      = (const float*)d_in[0];
    const float* wqkv   = (const float*)d_in[1];
    const float* wo     = (const float*)d_in[2];
    const float* w1     = (const float*)d_in[3];
    const float* w2     = (const float*)d_in[4];
    const float* w3     = (const float*)d_in[5];
    const float* anw    = (const float*)d_in[6];
    const float* fnw    = (const float*)d_in[7];
    const float* k_cache= (const float*)d_in[8];
    const float* v_cache= (const float*)d_in[9];
    const float* k_lcache=(const float*)d_in[10];
    const float* freqs  = (const float*)d_in[11];
    const float* mask1  = (const float*)d_in[12];
    const int*   sorted = (const int*)d_in[13];
    const int*   pos_p  = (const int*)d_in[14];
    // d_in[15] = heavy_const (256, fixed at compile time here)

    float* ws    = (float*)d_ws;
    float* xn    = ws;                    // 4*4096
    float* qkv   = xn    + BATCH*DIM;     // 4*12288
    float* q_rot = qkv   + BATCH*QKVDIM;  // 4*4096
    float* k_rot = q_rot + BATCH*DIM;     // 4*4096
    float* q_lab = k_rot + BATCH*DIM;     // 4*1024
    float* k_lab = q_lab + BATCH*NH*HC;   // 4*1024
    float* o_buf = k_lab + BATCH*NH*HC;   // 4*4096
    float* h_buf = o_buf + BATCH*DIM;     // 4*4096
    float* hn    = h_buf + BATCH*DIM;     // 4*4096
    float* act   = hn    + BATCH*DIM;     // 4*11008
    float* outf  = (float*)d_out;

    rmsnorm_kernel<<<BATCH, 256, 0, stream>>>(x, anw, xn);
    gemv_wmma_kernel<256><<<QKVDIM/16, 256, 0, stream>>>(xn, wqkv, qkv, nullptr, DIM, QKVDIM);
    rope_kernel<<<BATCH*NH, 128, 0, stream>>>(qkv, freqs, sorted, q_rot, k_rot, q_lab, k_lab);
    attn_topk_kernel<<<BATCH*NH, 256, 0, stream>>>(q_rot, k_rot, qkv, q_lab, k_lab,
                                                   k_lcache, mask1, k_cache, v_cache,
                                                   pos_p, o_buf);
    gemv_wmma_kernel<256><<<DIM/16, 256, 0, stream>>>(o_buf, wo, h_buf, x, DIM, DIM);
    rmsnorm_kernel<<<BATCH, 256, 0, stream>>>(h_buf, fnw, hn);
    gemv2_silu_kernel<128><<<INTER/16, 256, 0, stream>>>(hn, w1, w3, act, DIM, INTER);
    gemv_wmma_kernel<256><<<DIM/16, 256, 0, stream>>>(act, w2, outf, h_buf, INTER, DIM);
}